// HierarchicalRNN_55078660604133
// MI455X (gfx1250) — compile-verified
//
#include <hip/hip_runtime.h>
#include <hip/hip_bf16.h>

typedef __attribute__((ext_vector_type(2))) float v2f;
typedef __attribute__((ext_vector_type(8))) float v8f;

#define T_ 32
#define B_ 16
#define I_ 128
#define H_ 384
#define O_ 32
#define E_ 307      // int(0.8*384) == 384 - round(0.2*384)
#define AXc 0.2f
#define AWc 0.02f
#define DTc 0.02f

__device__ __forceinline__ float relu_(float x) { return fmaxf(x, 0.f); }
__device__ __forceinline__ float retanh_(float x) { return tanhf(fmaxf(x, 0.f)); }

// MODE 0: all-ones (x2c). MODE 1: EI signed, diag removed (c2c, h2h).
// MODE 2: excitatory-truncated (c2h, h2o, attn): +1 for j<307 else 0.
template <int MODE>
__device__ __forceinline__ float maskv(int o, int j) {
    if (MODE == 0) return 1.f;
    if (MODE == 1) return (j == o) ? 0.f : ((j < E_) ? 1.f : -1.f);
    return (j < E_) ? 1.f : 0.f;
}

// Precompute Weff = relu(W) * mask once per launch (time-invariant).
template <int MODE>
__global__ void k_prep(const float* __restrict__ W, float* __restrict__ Weff, int rows,
                       int cols) {
    const int idx = blockIdx.x * blockDim.x + threadIdx.x;
    if (idx >= rows * cols) return;
    const int o = idx / cols;
    const int j = idx - o * cols;
    Weff[idx] = relu_(W[idx]) * maskv<MODE>(o, j);
}

// One wave computes a 16(batch) x 16(out-row) f32 tile via V_WMMA_F32_16X16X4_F32.
// A frag: 16x4 f32 (lane<16: M=lane, K=k0+{0,1}; lane>=16: M=lane-16, K=k0+{2,3})
// B frag: 4x16 f32 (lane<16: N=lane, K=k0+{0,1}; lane>=16: N=lane-16, K=k0+{2,3})
// Weff is the precomputed relu(W)*mask, row-major [Hout, K].
__device__ __forceinline__ v8f wmma_gemm(v8f acc, const float* __restrict__ inp,
                                         const float* __restrict__ Weff, int K, int o0,
                                         int lane) {
    if (inp == nullptr) return acc;  // wave-uniform: zero contribution
    const int half = lane >> 4, lr = lane & 15;
    const float* arow = inp + lr * K;
    const float* brow = Weff + (o0 + lr) * K;
    for (int k0 = 0; k0 < K; k0 += 4) {
        const int ka = k0 + 2 * half;
        const v2f a = *(const v2f*)(arow + ka);
        const v2f b = *(const v2f*)(brow + ka);
        acc = __builtin_amdgcn_wmma_f32_16x16x4_f32(false, a, false, b, (short)0, acc,
                                                    false, false);
    }
    return acc;
}

// ---- Shared-weight GEMMs, cortex ----
__global__ void k_gemm_c(const float* __restrict__ x_t, const float* __restrict__ noc_tm1,
                         const float* __restrict__ noh_tm1, const float* __restrict__ Ex2c,
                         const float* __restrict__ Ec2c, const float* __restrict__ b_c2c,
                         const float* __restrict__ Eattn, const float* __restrict__ b_attn,
                         float* __restrict__ tic_shared) {
    const int lane = threadIdx.x;
    const int o0 = blockIdx.x * 16;
    v8f accA = {};
    accA = wmma_gemm(accA, noh_tm1, Eattn, H_, o0, lane);
    v8f accM = {};
    accM = wmma_gemm(accM, x_t, Ex2c, I_, o0, lane);
    accM = wmma_gemm(accM, noc_tm1, Ec2c, H_, o0, lane);
    const int half = lane >> 4, lr = lane & 15;
    const int o = o0 + lr;
#pragma unroll
    for (int r = 0; r < 8; r++) {
        const int b = r + 8 * half;
        const float attn = retanh_(accA[r] + b_attn[o]);
        tic_shared[b * H_ + o] = accM[r] + b_c2c[o] + attn;
    }
}

// ---- Shared-weight GEMMs, hippocampus ----
__global__ void k_gemm_h(const float* __restrict__ noc_t, const float* __restrict__ noh_tm1,
                         const float* __restrict__ Ec2h, const float* __restrict__ Eh2h,
                         const float* __restrict__ b_h2h, float* __restrict__ tih_shared) {
    const int lane = threadIdx.x;
    const int o0 = blockIdx.x * 16;
    v8f acc = {};
    acc = wmma_gemm(acc, noc_t, Ec2h, H_, o0, lane);
    acc = wmma_gemm(acc, noh_tm1, Eh2h, H_, o0, lane);
    const int half = lane >> 4, lr = lane & 15;
    const int o = o0 + lr;
#pragma unroll
    for (int r = 0; r < 8; r++) {
        const int b = r + 8 * half;
        tih_shared[b * H_ + o] = acc[r] + b_h2h[o];
    }
}

// ---- Output projection ----
__global__ void k_out(const float* __restrict__ noh_t, const float* __restrict__ Eh2o,
                      float* __restrict__ out_t) {
    const int lane = threadIdx.x;
    const int o0 = blockIdx.x * 16;
    v8f acc = {};
    acc = wmma_gemm(acc, noh_t, Eh2o, H_, o0, lane);
    const int half = lane >> 4, lr = lane & 15;
#pragma unroll
    for (int r = 0; r < 8; r++) {
        const int b = r + 8 * half;
        out_t[b * O_ + o0 + lr] = acc[r];
    }
}

__device__ __forceinline__ float plast_upd(float wold, float Wv, float pre, float post,
                                           float R, float k0, float k1, float k2) {
    return fmaxf(wold * (1.f - AWc) + DTc * R * (k0 * pre + k1 * post + k2 * post * pre),
                 -fmaxf(Wv, 0.f));
}

// ---- Cortex fast-weight streaming (float4 / B128 path) ----
__global__ void k_cortex(const float* __restrict__ x, const float* __restrict__ Rs_tm1,
                         int t, int first, const float* __restrict__ noc_tm1,
                         const float* __restrict__ noc_tm2, const float* __restrict__ W_x2c,
                         const float* __restrict__ W_c2c, const float* __restrict__ kappa,
                         float* __restrict__ wxc, float* __restrict__ wcc,
                         const float* __restrict__ tic_shared, float* __restrict__ sc,
                         float* __restrict__ noc_out) {
    const int wave = (blockIdx.x * blockDim.x + threadIdx.x) >> 5;
    const int lane = threadIdx.x & 31;
    const int b = wave / H_;
    const int o = wave - b * H_;
    if (b >= B_) return;

    const float R = first ? 0.f : Rs_tm1[b];
    const float post = first ? 0.f : noc_tm1[b * H_ + o];
    const float k0x = kappa[0], k1x = kappa[1], k2x = kappa[2];
    const float k0c = kappa[3], k1c = kappa[4], k2c = kappa[5];
    const float4 z4 = {0.f, 0.f, 0.f, 0.f};

    float acc = 0.f;

    {   // wxc row (I=128): one float4 per lane
        const float* xt = x + (size_t)t * B_ * I_ + b * I_;
        const float* xp = first ? nullptr : x + (size_t)(t - 1) * B_ * I_ + b * I_;
        float* wrow = wxc + ((size_t)b * H_ + o) * I_;
        const float* Wr = W_x2c + (size_t)o * I_;
        const int i = lane * 4;
        const float4 Wv = *(const float4*)(Wr + i);
        const float4 wo = first ? z4 : *(const float4*)(wrow + i);
        const float4 pr = first ? z4 : *(const float4*)(xp + i);
        const float4 xv = *(const float4*)(xt + i);
        float4 nw;
        nw.x = plast_upd(wo.x, Wv.x, pr.x, post, R, k0x, k1x, k2x);
        nw.y = plast_upd(wo.y, Wv.y, pr.y, post, R, k0x, k1x, k2x);
        nw.z = plast_upd(wo.z, Wv.z, pr.z, post, R, k0x, k1x, k2x);
        nw.w = plast_upd(wo.w, Wv.w, pr.w, post, R, k0x, k1x, k2x);
        *(float4*)(wrow + i) = nw;
        acc += nw.x * xv.x + nw.y * xv.y + nw.z * xv.z + nw.w * xv.w;
    }

    {   // wcc row (H=384): 3 float4 iters
        float* wrow = wcc + ((size_t)b * H_ + o) * H_;
        const float* Wr = W_c2c + (size_t)o * H_;
        const float* prp = noc_tm2 ? noc_tm2 + b * H_ : nullptr;
        const float* ocp = noc_tm1 ? noc_tm1 + b * H_ : nullptr;
#pragma unroll
        for (int j0 = 0; j0 < H_; j0 += 128) {
            const int j = j0 + lane * 4;
            const float4 Wv = *(const float4*)(Wr + j);
            const float4 wo = first ? z4 : *(const float4*)(wrow + j);
            const float4 pr = prp ? *(const float4*)(prp + j) : z4;
            const float4 oc = ocp ? *(const float4*)(ocp + j) : z4;
            float4 nw;
            nw.x = plast_upd(wo.x, Wv.x, pr.x, post, R, k0c, k1c, k2c);
            nw.y = plast_upd(wo.y, Wv.y, pr.y, post, R, k0c, k1c, k2c);
            nw.z = plast_upd(wo.z, Wv.z, pr.z, post, R, k0c, k1c, k2c);
            nw.w = plast_upd(wo.w, Wv.w, pr.w, post, R, k0c, k1c, k2c);
            *(float4*)(wrow + j) = nw;
            const float mx = ((j + 0) == o) ? 0.f : (((j + 0) < E_) ? 1.f : -1.f);
            const float my = ((j + 1) == o) ? 0.f : (((j + 1) < E_) ? 1.f : -1.f);
            const float mz = ((j + 2) == o) ? 0.f : (((j + 2) < E_) ? 1.f : -1.f);
            const float mw = ((j + 3) == o) ? 0.f : (((j + 3) < E_) ? 1.f : -1.f);
            acc += nw.x * mx * oc.x + nw.y * my * oc.y + nw.z * mz * oc.z +
                   nw.w * mw * oc.w;
        }
    }

    for (int off = 16; off > 0; off >>= 1) acc += __shfl_xor(acc, off, 32);

    if (lane == 0) {
        const float tic = acc + tic_shared[b * H_ + o];
        const float scp = first ? 0.f : sc[b * H_ + o];
        const float nsc = scp * (1.f - AXc) + tic * AXc;
        sc[b * H_ + o] = nsc;
        noc_out[b * H_ + o] = retanh_(nsc);
    }
}

// ---- Hippocampus fast-weight streaming (float4 / B128 path) ----
__global__ void k_hippo(const float* __restrict__ Rs_tm1, int first,
                        const float* __restrict__ noc_t, const float* __restrict__ noc_tm1,
                        const float* __restrict__ noh_tm1, const float* __restrict__ noh_tm2,
                        const float* __restrict__ W_c2h, const float* __restrict__ W_h2h,
                        const float* __restrict__ kappa, float* __restrict__ wch,
                        float* __restrict__ whh, const float* __restrict__ tih_shared,
                        float* __restrict__ sh, float* __restrict__ noh_out) {
    const int wave = (blockIdx.x * blockDim.x + threadIdx.x) >> 5;
    const int lane = threadIdx.x & 31;
    const int b = wave / H_;
    const int o = wave - b * H_;
    if (b >= B_) return;

    const float R = first ? 0.f : Rs_tm1[b];
    const float post = first ? 0.f : noh_tm1[b * H_ + o];
    const float k0a = kappa[6], k1a = kappa[7], k2a = kappa[8];
    const float k0b = kappa[9], k1b = kappa[10], k2b = kappa[11];
    const float4 z4 = {0.f, 0.f, 0.f, 0.f};

    float acc = 0.f;

    {   // wch row: pre=noc_{t-1}, post=noh_{t-1}; input noc_t, mask j<E
        float* wrow = wch + ((size_t)b * H_ + o) * H_;
        const float* Wr = W_c2h + (size_t)o * H_;
        const float* prp = noc_tm1 ? noc_tm1 + b * H_ : nullptr;
        const float* inp = noc_t + b * H_;
#pragma unroll
        for (int j0 = 0; j0 < H_; j0 += 128) {
            const int j = j0 + lane * 4;
            const float4 Wv = *(const float4*)(Wr + j);
            const float4 wo = first ? z4 : *(const float4*)(wrow + j);
            const float4 pr = prp ? *(const float4*)(prp + j) : z4;
            const float4 in = *(const float4*)(inp + j);
            float4 nw;
            nw.x = plast_upd(wo.x, Wv.x, pr.x, post, R, k0a, k1a, k2a);
            nw.y = plast_upd(wo.y, Wv.y, pr.y, post, R, k0a, k1a, k2a);
            nw.z = plast_upd(wo.z, Wv.z, pr.z, post, R, k0a, k1a, k2a);
            nw.w = plast_upd(wo.w, Wv.w, pr.w, post, R, k0a, k1a, k2a);
            *(float4*)(wrow + j) = nw;
            const float mx = ((j + 0) < E_) ? 1.f : 0.f;
            const float my = ((j + 1) < E_) ? 1.f : 0.f;
            const float mz = ((j + 2) < E_) ? 1.f : 0.f;
            const float mw = ((j + 3) < E_) ? 1.f : 0.f;
            acc += nw.x * mx * in.x + nw.y * my * in.y + nw.z * mz * in.z +
                   nw.w * mw * in.w;
        }
    }

    {   // whh row: pre=noh_{t-2}, post=noh_{t-1}; input oh_t=noh_{t-1}, EI mask
        float* wrow = whh + ((size_t)b * H_ + o) * H_;
        const float* Wr = W_h2h + (size_t)o * H_;
        const float* prp = noh_tm2 ? noh_tm2 + b * H_ : nullptr;
        const float* ohp = noh_tm1 ? noh_tm1 + b * H_ : nullptr;
#pragma unroll
        for (int j0 = 0; j0 < H_; j0 += 128) {
            const int j = j0 + lane * 4;
            const float4 Wv = *(const float4*)(Wr + j);
            const float4 wo = first ? z4 : *(const float4*)(wrow + j);
            const float4 pr = prp ? *(const float4*)(prp + j) : z4;
            const float4 oh = ohp ? *(const float4*)(ohp + j) : z4;
            float4 nw;
            nw.x = plast_upd(wo.x, Wv.x, pr.x, post, R, k0b, k1b, k2b);
            nw.y = plast_upd(wo.y, Wv.y, pr.y, post, R, k0b, k1b, k2b);
            nw.z = plast_upd(wo.z, Wv.z, pr.z, post, R, k0b, k1b, k2b);
            nw.w = plast_upd(wo.w, Wv.w, pr.w, post, R, k0b, k1b, k2b);
            *(float4*)(wrow + j) = nw;
            const float mx = ((j + 0) == o) ? 0.f : (((j + 0) < E_) ? 1.f : -1.f);
            const float my = ((j + 1) == o) ? 0.f : (((j + 1) < E_) ? 1.f : -1.f);
            const float mz = ((j + 2) == o) ? 0.f : (((j + 2) < E_) ? 1.f : -1.f);
            const float mw = ((j + 3) == o) ? 0.f : (((j + 3) < E_) ? 1.f : -1.f);
            acc += nw.x * mx * oh.x + nw.y * my * oh.y + nw.z * mz * oh.z +
                   nw.w * mw * oh.w;
        }
    }

    for (int off = 16; off > 0; off >>= 1) acc += __shfl_xor(acc, off, 32);

    if (lane == 0) {
        const float tih = acc + tih_shared[b * H_ + o];
        const float shp = first ? 0.f : sh[b * H_ + o];
        const float nsh = shp * (1.f - AXc) + tih * AXc;
        sh[b * H_ + o] = nsh;
        noh_out[b * H_ + o] = retanh_(nsh);
    }
}

extern "C" void kernel_launch(void* const* d_in, const int* in_sizes, int n_in,
                              void* d_out, int out_size, void* d_ws, size_t ws_size,
                              hipStream_t stream) {
    const float* x = (const float*)d_in[0];
    const float* Rs = (const float*)d_in[1];
    const float* W_x2c = (const float*)d_in[2];
    const float* W_c2c = (const float*)d_in[3];
    const float* b_c2c = (const float*)d_in[4];
    const float* W_c2h = (const float*)d_in[5];
    const float* W_h2h = (const float*)d_in[6];
    const float* b_h2h = (const float*)d_in[7];
    const float* W_h2o = (const float*)d_in[8];
    const float* W_attn = (const float*)d_in[9];
    const float* b_attn = (const float*)d_in[10];
    const float* kappa = (const float*)d_in[11];
    float* out = (float*)d_out;

    float* ws = (float*)d_ws;
    size_t off = 0;
    float* wxc = ws + off; off += (size_t)B_ * H_ * I_;
    float* wcc = ws + off; off += (size_t)B_ * H_ * H_;
    float* wch = ws + off; off += (size_t)B_ * H_ * H_;
    float* whh = ws + off; off += (size_t)B_ * H_ * H_;
    float* noc_hist = ws + off; off += (size_t)T_ * B_ * H_;
    float* noh_hist = ws + off; off += (size_t)T_ * B_ * H_;
    float* sc = ws + off; off += (size_t)B_ * H_;
    float* sh = ws + off; off += (size_t)B_ * H_;
    float* tic_s = ws + off; off += (size_t)B_ * H_;
    float* tih_s = ws + off; off += (size_t)B_ * H_;
    float* Ex2c = ws + off; off += (size_t)H_ * I_;
    float* Ec2c = ws + off; off += (size_t)H_ * H_;
    float* Ec2h = ws + off; off += (size_t)H_ * H_;
    float* Eh2h = ws + off; off += (size_t)H_ * H_;
    float* Eattn = ws + off; off += (size_t)H_ * H_;
    float* Eh2o = ws + off; off += (size_t)O_ * H_;

    // Precompute relu(W)*mask once per launch (time-invariant across 32 steps).
    k_prep<0><<<(H_ * I_ + 255) / 256, 256, 0, stream>>>(W_x2c, Ex2c, H_, I_);
    k_prep<1><<<(H_ * H_ + 255) / 256, 256, 0, stream>>>(W_c2c, Ec2c, H_, H_);
    k_prep<2><<<(H_ * H_ + 255) / 256, 256, 0, stream>>>(W_c2h, Ec2h, H_, H_);
    k_prep<1><<<(H_ * H_ + 255) / 256, 256, 0, stream>>>(W_h2h, Eh2h, H_, H_);
    k_prep<2><<<(H_ * H_ + 255) / 256, 256, 0, stream>>>(W_attn, Eattn, H_, H_);
    k_prep<2><<<(O_ * H_ + 255) / 256, 256, 0, stream>>>(W_h2o, Eh2o, O_, H_);

    const int stream_blocks = (B_ * H_) / 8;  // 8 waves (256 threads) per block

    for (int t = 0; t < T_; t++) {
        const int first = (t == 0) ? 1 : 0;
        const float* Rp = first ? nullptr : Rs + (size_t)(t - 1) * B_;
        const float* noc_tm1 = (t >= 1) ? noc_hist + (size_t)(t - 1) * B_ * H_ : nullptr;
        const float* noc_tm2 = (t >= 2) ? noc_hist + (size_t)(t - 2) * B_ * H_ : nullptr;
        const float* noh_tm1 = (t >= 1) ? noh_hist + (size_t)(t - 1) * B_ * H_ : nullptr;
        const float* noh_tm2 = (t >= 2) ? noh_hist + (size_t)(t - 2) * B_ * H_ : nullptr;
        float* noc_t = noc_hist + (size_t)t * B_ * H_;
        float* noh_t = noh_hist + (size_t)t * B_ * H_;
        const float* x_t = x + (size_t)t * B_ * I_;

        k_gemm_c<<<H_ / 16, 32, 0, stream>>>(x_t, noc_tm1, noh_tm1, Ex2c, Ec2c, b_c2c,
                                             Eattn, b_attn, tic_s);
        k_cortex<<<stream_blocks, 256, 0, stream>>>(x, Rp, t, first, noc_tm1, noc_tm2,
                                                    W_x2c, W_c2c, kappa, wxc, wcc, tic_s,
                                                    sc, noc_t);
        k_gemm_h<<<H_ / 16, 32, 0, stream>>>(noc_t, noh_tm1, Ec2h, Eh2h, b_h2h, tih_s);
        k_hippo<<<stream_blocks, 256, 0, stream>>>(Rp, first, noc_t, noc_tm1, noh_tm1,
                                                   noh_tm2, W_c2h, W_h2h, kappa, wch, whh,
                                                   tih_s, sh, noh_t);
        k_out<<<O_ / 16, 32, 0, stream>>>(noh_t, Eh2o, out + (size_t)t * B_ * O_);
    }
}